// RiemannianGeodesicDistance_89876485636497
// MI455X (gfx1250) — compile-verified
//
#include <hip/hip_runtime.h>

#define NN      64
#define ST      68          // LDS row stride (floats); 68 % 64 == 4 -> conflict-free column walks
#define BATCH   256
#define NPROTO  32
#define SWEEPS1 12
#define SWEEPS2 10

typedef __attribute__((ext_vector_type(2))) float v2f;
typedef __attribute__((ext_vector_type(8))) float v8f;

// ---------------------------------------------------------------------------
// 64x64 f32 GEMM on one workgroup (8 wave32) using V_WMMA_F32_16X16X4_F32.
// A, B are LDS tiles with row stride ST. Each wave computes two 16x16 output
// tiles, accumulating over K in steps of 4. Compile-time specialization:
//   HAS_SCALE: fold per-K diag scale into the A fragment (V * diag(rs))
//   TRANSB:    read B transposed (for * V^T)
// C may be LDS (cstride=ST) or global (cstride=64).
// A-fragment (16x4 f32): lane m=l&15, hi=l>>4 -> a0=A[m][k+2*hi], a1=A[m][k+2*hi+1]
// B-fragment (4x16 f32): b0=B[k+2*hi][n],  b1=B[k+2*hi+1][n],  n=l&15
// C/D (16x16 f32): vgpr v, lane l -> row v+8*(l>>4), col l&15
// ---------------------------------------------------------------------------
template<bool HAS_SCALE, bool TRANSB>
__device__ __forceinline__ void mm64_wmma(const float* __restrict__ A,
                                          const float* __restrict__ B,
                                          const float* __restrict__ scaleK,
                                          float* __restrict__ Cdst, int cstride)
{
    const int lane = threadIdx.x & 31;
    const int wv   = threadIdx.x >> 5;     // 0..7
    const int m    = lane & 15;
    const int hi   = lane >> 4;
#pragma unroll
    for (int t = 0; t < 2; ++t) {
        const int tile = wv * 2 + t;       // 0..15
        const int tr   = (tile >> 2) << 4; // tile row base
        const int tc   = (tile &  3) << 4; // tile col base
        v8f acc = {};
#pragma unroll
        for (int k = 0; k < NN; k += 4) {
            const int kk = k + 2 * hi;
            v2f a, bb;
            a[0] = A[(tr + m) * ST + kk];
            a[1] = A[(tr + m) * ST + kk + 1];
            if (HAS_SCALE) {
                a[0] *= scaleK[kk];
                a[1] *= scaleK[kk + 1];
            }
            if (TRANSB) {
                bb[0] = B[(tc + m) * ST + kk];
                bb[1] = B[(tc + m) * ST + kk + 1];
            } else {
                bb[0] = B[kk * ST + tc + m];
                bb[1] = B[(kk + 1) * ST + tc + m];
            }
            acc = __builtin_amdgcn_wmma_f32_16x16x4_f32(
                      false, a, false, bb, (short)0, acc, false, false);
        }
#pragma unroll
        for (int v = 0; v < 8; ++v)
            Cdst[(tr + v + 8 * hi) * cstride + tc + m] = acc[v];
    }
}

// ---------------------------------------------------------------------------
// One parallel-Jacobi round: round-robin pairing of 64 indices into 32
// disjoint (p,q) rotations, applied as A <- J^T A J (and V <- V J if WITH_V).
// 256 threads; each thread owns 16 contiguous elements of a row (row pass)
// or of a column (column pass). Read-sync-write keeps it in place.
// ---------------------------------------------------------------------------
template<bool WITH_V>
__device__ __forceinline__ void jacobi_round(float* __restrict__ A,
                                             float* __restrict__ V,
                                             int r,
                                             float* __restrict__ g1,
                                             float* __restrict__ g2,
                                             int*   __restrict__ pr)
{
    const int t = threadIdx.x;

    if (t < 32) {
        const int p = (t == 0) ? 0 : 1 + ((t - 1 + r) % 63);
        const int q = 1 + ((62 - t + r) % 63);
        const float app = A[p * ST + p];
        const float aqq = A[q * ST + q];
        const float apq = A[p * ST + q];
        float c = 1.0f, s = 0.0f;
        if (__builtin_fabsf(apq) > 1e-12f) {
            const float tau = (aqq - app) / (2.0f * apq);
            const float tt  = (tau >= 0.0f ? 1.0f : -1.0f) /
                              (__builtin_fabsf(tau) + __builtin_sqrtf(1.0f + tau * tau));
            c = __frsqrt_rn(1.0f + tt * tt);
            s = tt * c;
        }
        g1[p] = c; g2[p] = s;  pr[p] = q;
        g1[q] = c; g2[q] = -s; pr[q] = p;
    }
    __syncthreads();

    // Row pass: row'_j = g1[j]*row_j + g2[j]*row_partner(j)
    {
        const int row = t >> 2, x0 = (t & 3) << 4;
        const int pj = pr[row];
        const float a1 = g1[row], a2 = g2[row];
        float sv[16], pv[16];
#pragma unroll
        for (int i = 0; i < 16; ++i) {
            sv[i] = A[row * ST + x0 + i];
            pv[i] = A[pj  * ST + x0 + i];
        }
        __syncthreads();
#pragma unroll
        for (int i = 0; i < 16; ++i)
            A[row * ST + x0 + i] = a1 * sv[i] + a2 * pv[i];
    }
    __syncthreads();

    // Column pass on A
    {
        const int col = t >> 2, x0 = (t & 3) << 4;
        const int pj = pr[col];
        const float a1 = g1[col], a2 = g2[col];
        float sv[16], pv[16];
#pragma unroll
        for (int i = 0; i < 16; ++i) {
            sv[i] = A[(x0 + i) * ST + col];
            pv[i] = A[(x0 + i) * ST + pj];
        }
        __syncthreads();
#pragma unroll
        for (int i = 0; i < 16; ++i)
            A[(x0 + i) * ST + col] = a1 * sv[i] + a2 * pv[i];
    }
    __syncthreads();

    // Column pass on V (eigenvector accumulation), compile-time optional
    if (WITH_V) {
        const int col = t >> 2, x0 = (t & 3) << 4;
        const int pj = pr[col];
        const float a1 = g1[col], a2 = g2[col];
        float sv[16], pv[16];
#pragma unroll
        for (int i = 0; i < 16; ++i) {
            sv[i] = V[(x0 + i) * ST + col];
            pv[i] = V[(x0 + i) * ST + pj];
        }
        __syncthreads();
#pragma unroll
        for (int i = 0; i < 16; ++i)
            V[(x0 + i) * ST + col] = a1 * sv[i] + a2 * pv[i];
        __syncthreads();
    }
}

// ---------------------------------------------------------------------------
// Kernel 1: per-b eigendecomposition of x_b, emit invS_b = V diag(w^-1/2) V^T
// ---------------------------------------------------------------------------
__global__ __launch_bounds__(256)
void spd_invsqrt_kernel(const float* __restrict__ x, float* __restrict__ invS)
{
    __shared__ float sA[NN * ST];
    __shared__ float sV[NN * ST];
    __shared__ float g1[NN], g2[NN], rs[NN];
    __shared__ int   pr[NN];

    const int b = blockIdx.x;
    const float* X = x + (size_t)b * NN * NN;

    for (int e = threadIdx.x; e < NN * NN; e += 256) {
        const int i = e >> 6, j = e & 63;
        sA[i * ST + j] = X[e];
        sV[i * ST + j] = (i == j) ? 1.0f : 0.0f;
    }
    __syncthreads();

    for (int sweep = 0; sweep < SWEEPS1; ++sweep)
        for (int r = 0; r < 63; ++r)
            jacobi_round<true>(sA, sV, r, g1, g2, pr);

    if (threadIdx.x < NN) {
        float w = sA[threadIdx.x * ST + threadIdx.x];
        w = fmaxf(w, 1e-30f);
        rs[threadIdx.x] = __frsqrt_rn(w);
    }
    __syncthreads();

    // invS = (V * diag(rs)) * V^T  -> straight to global workspace (row stride 64)
    mm64_wmma<true, true>(sV, sV, rs, invS + (size_t)b * NN * NN, NN);
}

// ---------------------------------------------------------------------------
// Kernel 2: per (b,c): M = S P S, eigenvalues-only Jacobi, sum (log w / w)^2
// ---------------------------------------------------------------------------
__global__ __launch_bounds__(256)
void geodist_kernel(const float* __restrict__ protos,
                    const float* __restrict__ invS,
                    float* __restrict__ out)
{
    __shared__ float sA[NN * ST];   // S = x_b^{-1/2}
    __shared__ float sB[NN * ST];   // P_c, then M
    __shared__ float sC[NN * ST];   // T1 = S P
    __shared__ float g1[NN], g2[NN];
    __shared__ int   pr[NN];

    const int c = blockIdx.x;
    const int b = blockIdx.y;
    const float* S = invS   + (size_t)b * NN * NN;
    const float* P = protos + (size_t)c * NN * NN;

    for (int e = threadIdx.x; e < NN * NN; e += 256) {
        const int i = e >> 6, j = e & 63;
        sA[i * ST + j] = S[e];
        sB[i * ST + j] = P[e];
    }
    __syncthreads();

    mm64_wmma<false, false>(sA, sB, nullptr, sC, ST);   // T1 = S P
    __syncthreads();
    mm64_wmma<false, false>(sC, sA, nullptr, sB, ST);   // M  = T1 S  (overwrites P)
    __syncthreads();

    // Symmetrize M (eigh-stability, matches reference)
    {
        const int row = threadIdx.x >> 2, x0 = (threadIdx.x & 3) << 4;
        float a[16], bt[16];
#pragma unroll
        for (int i = 0; i < 16; ++i) {
            a[i]  = sB[row * ST + x0 + i];
            bt[i] = sB[(x0 + i) * ST + row];
        }
        __syncthreads();
#pragma unroll
        for (int i = 0; i < 16; ++i)
            sB[row * ST + x0 + i] = 0.5f * (a[i] + bt[i]);
        __syncthreads();
    }

    for (int sweep = 0; sweep < SWEEPS2; ++sweep)
        for (int r = 0; r < 63; ++r)
            jacobi_round<false>(sB, nullptr, r, g1, g2, pr);

    // dists = sum_i (log w_i / w_i)^2   (exact reduction of tr((P^-1 U)^2))
    if (threadIdx.x < NN) {
        float w = fmaxf(sB[threadIdx.x * ST + threadIdx.x], 1e-30f);
        const float v = __logf(w) / w;
        g1[threadIdx.x] = v * v;
    }
    __syncthreads();
    if (threadIdx.x == 0) {
        float acc = 0.0f;
#pragma unroll
        for (int i = 0; i < NN; ++i) acc += g1[i];
        out[(size_t)b * NPROTO + c] = acc;
    }
}

// ---------------------------------------------------------------------------
extern "C" void kernel_launch(void* const* d_in, const int* in_sizes, int n_in,
                              void* d_out, int out_size, void* d_ws, size_t ws_size,
                              hipStream_t stream)
{
    (void)in_sizes; (void)n_in; (void)out_size; (void)ws_size;
    const float* x      = (const float*)d_in[0];   // (256, 64, 64)
    const float* protos = (const float*)d_in[1];   // (32, 64, 64)
    float*       out    = (float*)d_out;           // (256, 32)
    float*       invS   = (float*)d_ws;            // 256*4096 floats = 4 MB

    spd_invsqrt_kernel<<<dim3(BATCH), dim3(256), 0, stream>>>(x, invS);
    geodist_kernel<<<dim3(NPROTO, BATCH), dim3(256), 0, stream>>>(protos, invS, out);
}